// NearestEmbed_13864154431909
// MI455X (gfx1250) — compile-verified
//
#include <hip/hip_runtime.h>

// ---------------------------------------------------------------------------
// NearestEmbed (VQ) for MI455X / gfx1250, wave32 + V_WMMA_F32_16X16X4_F32.
//
// argmin_k ||x - w_k||^2  ==  argmax_k ( x.w_k - 0.5*||w_k||^2 )
// Accumulator C is preloaded with -0.5*||w_k||^2; the dot product accumulates
// through 16 chained fp32 WMMAs over d=64 (fp32 kept for argmin robustness:
// weights are O(1/512), so margins are tiny).
//
// Weight is staged into LDS pre-swizzled into the WMMA B-operand layout so the
// hot loop issues ONE conflict-free ds_load_b64 per B tile with an immediate
// offset (b64 across 32 lanes touches all 64 banks exactly once).
// ---------------------------------------------------------------------------

typedef float v2f __attribute__((ext_vector_type(2)));
typedef float v8f __attribute__((ext_vector_type(8)));

#define D_DIM        64
#define K_CODES      512
#define N_SPATIAL    1024                  // 32*32
#define BATCH        64
#define M_TOTAL      (BATCH * N_SPATIAL)   // 65536 rows
#define WAVES_PER_BLOCK 8
#define TILES_PER_WAVE  2                  // 2 x 16-row M tiles per wave: B reuse x2
#define ROWS_PER_BLOCK  (WAVES_PER_BLOCK * TILES_PER_WAVE * 16)  // 256

// Swizzled-B dword index for weight element (d, k):
//   kt=k>>4, kcol=k&15, s=d>>2, half=(d>>1)&1, e=d&1, lane=half*16+kcol
//   idx = ((kt*16 + s)*32 + lane)*2 + e
__device__ __forceinline__ int b_swizzle_idx(int d, int k) {
    const int kt   = k >> 4;
    const int kcol = k & 15;
    const int s    = d >> 2;
    const int half = (d >> 1) & 1;
    const int e    = d & 1;
    return (((kt * 16 + s) * 32) + half * 16 + kcol) * 2 + e;
}

extern "C" __global__ __launch_bounds__(256)
void nearest_embed_wmma(const float* __restrict__ x,
                        const float* __restrict__ w,
                        float* __restrict__ out)
{
    extern __shared__ float lds[];
    float* lb = lds;                           // [32768] weight, WMMA-B swizzled
    float* lc = lds + D_DIM * K_CODES;         // [512]  -0.5*||w_k||^2
    int*   lk = (int*)(lc + K_CODES);          // [256]  best code per row

    const int tid  = threadIdx.x;
    const int lane = tid & 31;
    const int wv   = tid >> 5;
    const int half = lane >> 4;                // 0: lanes 0-15, 1: lanes 16-31
    const int lc16 = lane & 15;

    // ---- stage weight into LDS in WMMA B-operand order (coalesced global) ----
    for (int i = tid; i < D_DIM * K_CODES; i += 256) {
        const int d = i >> 9;                  // i = d*512 + k
        const int k = i & (K_CODES - 1);
        lb[b_swizzle_idx(d, k)] = w[i];
    }
    // ---- c[k] = -0.5 * sum_d w[d][k]^2 (from global, overlaps staging) ------
    for (int k = tid; k < K_CODES; k += 256) {
        float s = 0.f;
#pragma unroll
        for (int d = 0; d < D_DIM; ++d) {
            const float t = w[d * K_CODES + k];
            s += t * t;
        }
        lc[k] = -0.5f * s;
    }
    __syncthreads();

    // ---------------- per-wave GEMM + argmax ---------------------------------
    const int m0 = blockIdx.x * ROWS_PER_BLOCK + wv * (TILES_PER_WAVE * 16);
    const int b  = m0 / N_SPATIAL;             // constant per block (256 | 1024)
    const int n0 = m0 % N_SPATIAL;
    const float* xb = x + (size_t)b * D_DIM * N_SPATIAL;

    // A preload: lane = row (n); lane-half + vgpr pair = d (k0/k1 | k2/k3).
    v2f a[TILES_PER_WAVE][16];
#pragma unroll
    for (int t = 0; t < TILES_PER_WAVE; ++t) {
        const int n = n0 + t * 16 + lc16;
#pragma unroll
        for (int s = 0; s < 16; ++s) {
            const int d0 = s * 4 + half * 2;
            a[t][s].x = xb[(size_t)d0       * N_SPATIAL + n];
            a[t][s].y = xb[(size_t)(d0 + 1) * N_SPATIAL + n];
        }
    }

    float bestS[TILES_PER_WAVE][8];
    int   bestI[TILES_PER_WAVE][8];
#pragma unroll
    for (int t = 0; t < TILES_PER_WAVE; ++t)
#pragma unroll
        for (int v = 0; v < 8; ++v) { bestS[t][v] = -INFINITY; bestI[t][v] = 0; }

    // Lane's B slot for tile (kt, s) starts at dword (kt*512 + s*32 + lane)*2.
    const v2f* lbl = (const v2f*)lb + lane;    // + kt*512 + s*32 below

    for (int kt = 0; kt < K_CODES / 16; ++kt) {
        const int   kcol  = kt * 16 + lc16;    // this lane's column (all C vgprs)
        const float cinit = lc[kcol];
        v8f acc0 = {cinit, cinit, cinit, cinit, cinit, cinit, cinit, cinit};
        v8f acc1 = acc0;
        const v2f* lbk = lbl + kt * 512;

#pragma unroll
        for (int s = 0; s < 16; ++s) {
            const v2f bb = lbk[s * 32];        // one ds_load_b64, imm offset s*256
            acc0 = __builtin_amdgcn_wmma_f32_16x16x4_f32(
                       false, a[0][s], false, bb, (short)0, acc0, false, false);
            acc1 = __builtin_amdgcn_wmma_f32_16x16x4_f32(
                       false, a[1][s], false, bb, (short)0, acc1, false, false);
        }

        // running argmax per lane (strict > keeps earliest == smallest k on ties)
#pragma unroll
        for (int v = 0; v < 8; ++v) {
            if (acc0[v] > bestS[0][v]) { bestS[0][v] = acc0[v]; bestI[0][v] = kcol; }
            if (acc1[v] > bestS[1][v]) { bestS[1][v] = acc1[v]; bestI[1][v] = kcol; }
        }
    }

    // cross-lane argmax within each 16-lane half (C: vgpr v -> rows v / v+8)
#pragma unroll
    for (int t = 0; t < TILES_PER_WAVE; ++t) {
#pragma unroll
        for (int v = 0; v < 8; ++v) {
            float s = bestS[t][v];
            int   i = bestI[t][v];
#pragma unroll
            for (int off = 8; off >= 1; off >>= 1) {
                const float os = __shfl_xor(s, off, 32);
                const int   oi = __shfl_xor(i, off, 32);
                if (os > s || (os == s && oi < i)) { s = os; i = oi; }
            }
            if (lc16 == 0) {   // lane 0 -> row v, lane 16 -> row v+8
                const int row = wv * (TILES_PER_WAVE * 16) + t * 16 + v + half * 8;
                lk[row] = i;
            }
        }
    }
    __syncthreads();

    // ---------------- coalesced output phase ---------------------------------
    // quantized: [B, D, N] floats, then argmin: [B, N] written as float values.
    const int n0b   = (blockIdx.x * ROWS_PER_BLOCK) % N_SPATIAL;
    const int myk   = lk[tid];
    const int nglob = n0b + tid;

    float* outq = out;
    float* outa = out + (size_t)BATCH * D_DIM * N_SPATIAL;
    outa[(size_t)b * N_SPATIAL + nglob] = (float)myk;

    // gather code vector from swizzled LDS; myk-dependent part hoisted
    const int kb = (myk >> 4) * 1024 + (myk & 15) * 2;
    float* oq = outq + (size_t)b * D_DIM * N_SPATIAL + nglob;
#pragma unroll 8
    for (int d = 0; d < D_DIM; ++d) {
        const int idx = kb + (d >> 2) * 64 + ((d >> 1) & 1) * 32 + (d & 1);
        oq[(size_t)d * N_SPATIAL] = lb[idx];
    }
}

extern "C" void kernel_launch(void* const* d_in, const int* in_sizes, int n_in,
                              void* d_out, int out_size, void* d_ws, size_t ws_size,
                              hipStream_t stream) {
    const float* x = (const float*)d_in[0];   // [64, 64, 32, 32] fp32
    const float* w = (const float*)d_in[1];   // [64, 512] fp32
    float* out = (float*)d_out;               // 64*64*1024 quantized + 64*1024 argmin

    const size_t lds_bytes = (size_t)(D_DIM * K_CODES + K_CODES) * sizeof(float)
                           + (size_t)ROWS_PER_BLOCK * sizeof(int);   // ~131 KB
    dim3 grid(M_TOTAL / ROWS_PER_BLOCK);   // 256 blocks
    dim3 block(256);                       // 8 waves (wave32)
    hipLaunchKernelGGL(nearest_embed_wmma, grid, block, lds_bytes, stream, x, w, out);
}